// YOLOLoss_14001593385146
// MI455X (gfx1250) — compile-verified
//
#include <hip/hip_runtime.h>

typedef float v2f __attribute__((ext_vector_type(2)));
typedef float v8f __attribute__((ext_vector_type(8)));

#define NBLK 256            // partial blocks per scale for obj and box/cls kernels
#define NT   256            // threads per block

// scale geometry (B=16, A=3, C=85, nc=80)
__device__ __forceinline__ int scaleH(int s)        { return 80 >> s; }          // 80,40,20
__device__ __forceinline__ int cellBase(int s)      { return s == 0 ? 0 : (s == 1 ? 102400 : 128000); }

__device__ __forceinline__ float bce(float x, float t) {
    return fmaxf(x, 0.0f) - x * t + log1pf(expf(-fabsf(x)));
}

__device__ __forceinline__ float ciou(float px, float py, float pw, float ph,
                                      float tx, float ty, float tw, float th) {
    const float EPS = 1e-7f;
    float px1 = px - pw * 0.5f, py1 = py - ph * 0.5f;
    float px2 = px + pw * 0.5f, py2 = py + ph * 0.5f;
    float tx1 = tx - tw * 0.5f, ty1 = ty - th * 0.5f;
    float tx2 = tx + tw * 0.5f, ty2 = ty + th * 0.5f;
    float iw = fmaxf(fminf(px2, tx2) - fmaxf(px1, tx1), 0.0f);
    float ih = fmaxf(fminf(py2, ty2) - fmaxf(py1, ty1), 0.0f);
    float inter = iw * ih;
    float uni   = pw * ph + tw * th - inter;
    float iou   = inter / (uni + EPS);
    float cd = (px - tx) * (px - tx) + (py - ty) * (py - ty);
    float ew = fmaxf(px2, tx2) - fminf(px1, tx1);
    float eh = fmaxf(py2, ty2) - fminf(py1, ty1);
    float diag = ew * ew + eh * eh + EPS;
    float dv = atanf(tw / th) - atanf(pw / ph);
    float v  = 0.40528473456935108577f * dv * dv;   // 4/pi^2
    float alpha = v / (1.0f - iou + v + EPS);
    return iou - cd / diag - alpha * v;
}

__device__ __forceinline__ float blockReduce256(float v, float* sm) {
    int t = threadIdx.x;
    sm[t] = v; __syncthreads();
    #pragma unroll
    for (int s = 128; s > 0; s >>= 1) {
        if (t < s) sm[t] += sm[t + s];
        __syncthreads();
    }
    float r = sm[0]; __syncthreads();
    return r;
}

// ---------------- kernel 1: zero the cell-winner grid + counters --------------
__global__ void yolo_zero(int* p, int n) {
    int i = blockIdx.x * blockDim.x + threadIdx.x;
    if (i < n) p[i] = 0;
}

// ---------------- kernel 2: scatter targets into cell grids ------------------
// blockIdx.x = scale. Deterministic duplicate resolution: max target index wins.
__global__ void yolo_assign(const float* __restrict__ targets, int nT,
                            int* __restrict__ winner, int* __restrict__ clsIdx,
                            float4* __restrict__ box, int* __restrict__ nCount) {
    int s = blockIdx.x;
    int H = scaleH(s);
    int base = cellBase(s);

    for (int t = threadIdx.x; t < nT; t += NT) {
        const float* tg = targets + t * 6;
        float bx = fminf(fmaxf(tg[2], 0.0f), 1.0f);
        float by = fminf(fmaxf(tg[3], 0.0f), 1.0f);
        int b  = (int)tg[0];
        int gx = min(max((int)(bx * (float)H), 0), H - 1);
        int gy = min(max((int)(by * (float)H), 0), H - 1);
        atomicMax(&winner[base + (b * H + gy) * H + gx], t + 1);
    }
    __threadfence();
    __syncthreads();
    for (int t = threadIdx.x; t < nT; t += NT) {
        const float* tg = targets + t * 6;
        float bx = fminf(fmaxf(tg[2], 0.0f), 1.0f);
        float by = fminf(fmaxf(tg[3], 0.0f), 1.0f);
        float bw = fminf(fmaxf(tg[4], 0.0f), 1.0f);
        float bh = fminf(fmaxf(tg[5], 0.0f), 1.0f);
        int b  = (int)tg[0];
        int c  = (int)tg[1];
        int gx = min(max((int)(bx * (float)H), 0), H - 1);
        int gy = min(max((int)(by * (float)H), 0), H - 1);
        int cell = base + (b * H + gy) * H + gx;
        if (winner[cell] == t + 1) {
            clsIdx[cell] = c;
            box[cell] = make_float4(bx, by, bw, bh);
            atomicAdd(&nCount[s], 1);
        }
    }
}

// ---------------- kernel 3: objectness BCE over all B*A*H*W cells ------------
// gridDim.x = 3*NBLK; strided channel-4 loads (only 1/85 of the tensor touched).
__global__ void yolo_obj(const float* __restrict__ p0, const float* __restrict__ p1,
                         const float* __restrict__ p2, const int* __restrict__ winner,
                         float* __restrict__ P) {
    __shared__ float sm[NT];
    int s   = blockIdx.x >> 8;
    int blk = blockIdx.x & (NBLK - 1);
    const float* pred = (s == 0) ? p0 : ((s == 1) ? p1 : p2);
    int H = scaleH(s), HW = H * H, base = cellBase(s);
    int N = 48 * HW;                           // B*A*H*W

    float acc = 0.0f;
    for (int e = blk * NT + threadIdx.x; e < N; e += NBLK * NT) {
        float x = pred[e * 85 + 4];
        int a = (e / HW) % 3;
        float t = 0.0f;
        if (a == 0) {
            int b  = e / (3 * HW);
            int hw = e % HW;
            t = (winner[base + b * HW + hw] > 0) ? 1.0f : 0.0f;
        }
        acc += bce(x, t);
    }
    float r = blockReduce256(acc, sm);
    if (threadIdx.x == 0) P[0 * 3 * NBLK + s * NBLK + blk] = r;
}

// ---------------- kernel 4: CIoU + class BCE on masked anchor-0 cells --------
__global__ void yolo_boxcls(const float* __restrict__ p0, const float* __restrict__ p1,
                            const float* __restrict__ p2, const int* __restrict__ winner,
                            const int* __restrict__ clsIdx, const float4* __restrict__ box,
                            float* __restrict__ P) {
    __shared__ float sm[NT];
    int s   = blockIdx.x >> 8;
    int blk = blockIdx.x & (NBLK - 1);
    const float* pred = (s == 0) ? p0 : ((s == 1) ? p1 : p2);
    int H = scaleH(s), HW = H * H, base = cellBase(s);
    int M = 16 * HW;                           // B*H*W

    float ciouAcc = 0.0f, clsAcc = 0.0f;
    for (int i = blk * NT + threadIdx.x; i < M; i += NBLK * NT) {
        if (winner[base + i] > 0) {
            int b  = i / HW;
            int hw = i % HW;
            int pb = ((b * 3) * HW + hw) * 85;  // anchor 0
            float4 tb = box[base + i];
            int c = clsIdx[base + i];
            ciouAcc += ciou(pred[pb + 0], pred[pb + 1], pred[pb + 2], pred[pb + 3],
                            tb.x, tb.y, tb.z, tb.w);
            #pragma unroll 4
            for (int j = 0; j < 80; j++)
                clsAcc += bce(pred[pb + 5 + j], (j == c) ? 1.0f : 0.0f);
        }
    }
    float rb = blockReduce256(ciouAcc, sm);
    float rc = blockReduce256(clsAcc, sm);
    if (threadIdx.x == 0) {
        P[1 * 3 * NBLK + s * NBLK + blk] = rb;
        P[2 * 3 * NBLK + s * NBLK + blk] = rc;
    }
}

// ---------------- kernel 5: final deterministic reduction via WMMA -----------
// Each of the 9 arrays (3 quantities x 3 scales) of 256 block partials is
// reduced: 8 values/lane in f32, then V_WMMA_F32_16X16X4_F32 with B=ones sums
// all 64 A elements into every D column (f32 precision preserved).
__global__ __launch_bounds__(32) void yolo_final(const float* __restrict__ P,
                                                 const int* __restrict__ nCount,
                                                 float* __restrict__ out) {
    int lane = threadIdx.x;
    float sums[9];
    #pragma unroll
    for (int i = 0; i < 9; i++) {
        const float* arr = P + i * NBLK;
        float p = 0.0f;
        #pragma unroll
        for (int j = 0; j < 8; j++) p += arr[lane * 8 + j];
        v2f a = {p, 0.0f};
        v2f bmat = {1.0f, 1.0f};
        v8f c = {};
        v8f d = __builtin_amdgcn_wmma_f32_16x16x4_f32(
            false, a, false, bmat, (short)0, c, false, false);
        float r = d[0] + d[1] + d[2] + d[3] + d[4] + d[5] + d[6] + d[7];
        // column N=0 of D lives in lanes 0 (M=0..7) and 16 (M=8..15)
        sums[i] = __shfl(r, 0, 32) + __shfl(r, 16, 32);
    }
    if (lane == 0) {
        const float Nfull[3] = {307200.0f, 76800.0f, 19200.0f};  // B*A*H*W
        float tb = 0.0f, to = 0.0f, tc = 0.0f;
        for (int s = 0; s < 3; s++) {
            float n = fmaxf((float)nCount[s], 1.0f);
            to += sums[0 * 3 + s] / Nfull[s];
            tb += 1.0f - sums[1 * 3 + s] / n;
            tc += sums[2 * 3 + s] / (n * 80.0f);
        }
        float total = 7.5f * tb + 1.0f * to + 0.5f * tc;
        out[0] = total; out[1] = tb; out[2] = to; out[3] = tc;
    }
}

extern "C" void kernel_launch(void* const* d_in, const int* in_sizes, int n_in,
                              void* d_out, int out_size, void* d_ws, size_t ws_size,
                              hipStream_t stream) {
    const float* p0      = (const float*)d_in[0];
    const float* p1      = (const float*)d_in[1];
    const float* p2      = (const float*)d_in[2];
    const float* targets = (const float*)d_in[3];
    int nT = in_sizes[3] / 6;

    // workspace layout (134400 cells total across scales)
    char* ws = (char*)d_ws;
    int*    winner = (int*)(ws);                    // 134400 ints
    int*    clsIdx = (int*)(ws + 537600);           // 134400 ints
    int*    nCount = (int*)(ws + 1075200);          // 4 ints
    float4* box    = (float4*)(ws + 1075216);       // 134400 float4
    float*  P      = (float*)(ws + 3225616);        // 9*256 partials
    float*  out    = (float*)d_out;

    // winner + clsIdx + nCount are the first 268804 ints (zero clsIdx harmlessly)
    yolo_zero<<<(268804 + NT - 1) / NT, NT, 0, stream>>>(winner, 268804);
    yolo_assign<<<3, NT, 0, stream>>>(targets, nT, winner, clsIdx, box, nCount);
    yolo_obj<<<3 * NBLK, NT, 0, stream>>>(p0, p1, p2, winner, P);
    yolo_boxcls<<<3 * NBLK, NT, 0, stream>>>(p0, p1, p2, winner, clsIdx, box, P);
    yolo_final<<<1, 32, 0, stream>>>(P, nCount, out);
}